// POLYPDE2D_30150670418420
// MI455X (gfx1250) — compile-verified
//
#include <hip/hip_runtime.h>

// ---------------------------------------------------------------------------
// PolyPDE2D on MI455X (gfx1250): 10 fused RHS passes (5 RK2-midpoint steps).
// Conv-as-GEMM on V_WMMA_F32_16X16X4_F32:
//   D[16 px x 16 feat] += A[16 px x 4 taps] * B[4 taps x 16 feat]
// taps t in [0,50): t<25 -> channel0 tap, t>=25 -> channel1; zero-padded to 52.
// feature n: channel n>>3, stencil n&7 (6 FD kernels + flipped 01/10 stencils).
// B matrix fully materialized (with zeros) in LDS -> branchless ds loads.
// A offsets are post-unroll compile-time constants selected by lane K-half.
// Poly/upwind stage is scalar VALU, 1 pixel per lane (wave32).
// ---------------------------------------------------------------------------

typedef float v2f __attribute__((ext_vector_type(2)));
typedef float v8f __attribute__((ext_vector_type(8)));

#define PDT 0.2f
#define PDX 0.0327249f

// LDS offset (in floats) of tap t inside s_in[2][20][20] for output pixel (0,0)
__device__ constexpr int aoff(int t) {
    int c  = t / 25;
    int rr = t - c * 25;
    return c * 400 + (rr / 5) * 20 + (rr % 5);
}

__global__ __launch_bounds__(256) void
polypde_rhs_step(const float* __restrict__ src,   // field fed into the conv
                 const float* ubase,              // field the update is added to
                 float* dst,                      // output field
                 const float* __restrict__ gk,    // raw kernels [6][5][5]
                 const float* __restrict__ pw0, const float* __restrict__ pb0,
                 const float* __restrict__ pw1, const float* __restrict__ pb1,
                 const float* __restrict__ pwf, const float* __restrict__ pbf,
                 float coef)
{
    __shared__ float s_in[2][20][20];     // 16x16 tile + 2-halo, both channels
    __shared__ float s_B[52][16];         // padded conv weight matrix [tap][feat]
    __shared__ float s_w0[48];            // [k][i][j] 2*2*12
    __shared__ float s_b0[4];
    __shared__ float s_w1[52];            // 2*2*13
    __shared__ float s_b1[4];
    __shared__ float s_wf[28];            // 2*14
    __shared__ float s_bf[2];
    __shared__ float s_D[8][2][16][20];   // per-wave D transpose scratch (padded)

    const int tid = threadIdx.x;
    const int b   = blockIdx.z;
    const int ty0 = blockIdx.y * 16;
    const int tx0 = blockIdx.x * 16;

    const float sc1 = 1.0f / PDX;         // dx^-1
    const float sc2 = sc1 * sc1;          // dx^-2

    // --- build padded B matrix: row = tap (52), col = feature (16).
    // col n: channel n>>3, stencil j=n&7. ks[j]=k[j]*dx^-ord for j<6;
    // j==6: -flipW(ks1); j==7: -flipH(ks2). Zero off-diagonal / padding rows.
    for (int t = tid; t < 832; t += 256) {
        int row = t >> 4, n = t & 15;
        float v = 0.0f;
        if (row < 50) {
            int tc = row >= 25;
            if (tc == (n >> 3)) {
                int rr = row - tc * 25, dy = rr / 5, dx = rr % 5;
                int j  = n & 7;
                if (j < 6) {
                    const float scl[6] = {1.0f, sc1, sc1, sc2, sc2, sc2};
                    v = gk[j * 25 + rr] * scl[j];
                } else if (j == 6) {
                    v = -gk[25 + dy * 5 + (4 - dx)] * sc1;
                } else {
                    v = -gk[50 + (4 - dy) * 5 + dx] * sc1;
                }
            }
        }
        s_B[row][n] = v;
    }
    if (tid < 48) s_w0[tid] = pw0[tid];
    if (tid < 4)  s_b0[tid] = pb0[tid];
    if (tid < 52) s_w1[tid] = pw1[tid];
    if (tid < 4)  s_b1[tid] = pb1[tid];
    if (tid < 28) s_wf[tid] = pwf[tid];
    if (tid < 2)  s_bf[tid] = pbf[tid];

    // --- cooperative periodic-halo load of both channels
    for (int t = tid; t < 800; t += 256) {
        int c = t / 400, r = t % 400, iy = r / 20, ix = r % 20;
        int gy = (ty0 + iy - 2 + 192) % 192;
        int gx = (tx0 + ix - 2 + 192) % 192;
        s_in[c][iy][ix] = src[(((size_t)b * 2 + c) * 192 + gy) * 192 + gx];
    }
    __syncthreads();

    const int wave = tid >> 5;
    const int lane = tid & 31;
    const int m    = lane & 15;   // A: pixel index / B,D: column N
    const int hf   = lane >> 4;   // K-half selector for A/B frags, M-half for D
    const int oy0  = 2 * wave;    // wave handles local output rows oy0, oy0+1

    const float* sinf = &s_in[0][0][0];
    const float* sbf  = &s_B[0][0];
    const int abase = oy0 * 20 + m;        // lane A base, row 0 (floats)
    const int bbase = (2 * hf) * 16 + m;   // lane B base (floats)

    v8f acc0 = {0.f, 0.f, 0.f, 0.f, 0.f, 0.f, 0.f, 0.f};
    v8f acc1 = acc0;

    // K = 52 -> 13 k-steps of 4; per ISA 16x4 A layout:
    // VGPR0 = K {4s+2*hf}, VGPR1 = K {4s+2*hf+1} across lane halves (B symmetric).
    #pragma unroll
    for (int s = 0; s < 13; ++s) {
        v2f bf_, af0, af1;
        bf_[0] = sbf[bbase + 64 * s];          // row 4s+2hf
        bf_[1] = sbf[bbase + 64 * s + 16];     // row 4s+2hf+1
        if (s < 12) {                          // all four taps < 50
            int oA0 = hf ? aoff(4 * s + 2) : aoff(4 * s + 0);
            int oA1 = hf ? aoff(4 * s + 3) : aoff(4 * s + 1);
            af0[0] = sinf[abase + oA0];       af0[1] = sinf[abase + oA1];
            af1[0] = sinf[abase + 20 + oA0];  af1[1] = sinf[abase + 20 + oA1];
        } else {                               // taps 48,49 valid; 50,51 -> 0
            float x0 = sinf[abase + aoff(48)],      x1 = sinf[abase + aoff(49)];
            float y0 = sinf[abase + 20 + aoff(48)], y1 = sinf[abase + 20 + aoff(49)];
            af0[0] = hf ? 0.f : x0;  af0[1] = hf ? 0.f : x1;
            af1[0] = hf ? 0.f : y0;  af1[1] = hf ? 0.f : y1;
        }
        acc0 = __builtin_amdgcn_wmma_f32_16x16x4_f32(
                   false, af0, false, bf_, (short)0, acc0, false, false);
        acc1 = __builtin_amdgcn_wmma_f32_16x16x4_f32(
                   false, af1, false, bf_, (short)0, acc1, false, false);
    }

    // --- transpose D through LDS: lane holds feature n=m for pixels i+8*hf
    #pragma unroll
    for (int i = 0; i < 8; ++i) {
        s_D[wave][0][i + 8 * hf][m] = acc0[i];
        s_D[wave][1][i + 8 * hf][m] = acc1[i];
    }
    __syncthreads();

    // --- poly + upwind: one pixel per lane (row hf, col m of wave's 2x16)
    float fe[16];
    #pragma unroll
    for (int n = 0; n < 16; ++n) fe[n] = s_D[wave][hf][m][n];

    float xv[12], xs[12];
    #pragma unroll
    for (int c = 0; c < 2; ++c)
        #pragma unroll
        for (int j = 0; j < 6; ++j) xv[c * 6 + j] = fe[c * 8 + j];
    #pragma unroll
    for (int j = 0; j < 12; ++j) xs[j] = xv[j];

    // upwind selection via analytic grad of poly_k at base
    #pragma unroll
    for (int k = 0; k < 2; ++k) {
        float o0a = s_b0[k * 2 + 0], o0b = s_b0[k * 2 + 1];
        #pragma unroll
        for (int j = 0; j < 12; ++j) {
            o0a += s_w0[k * 24 + j] * xv[j];
            o0b += s_w0[k * 24 + 12 + j] * xv[j];
        }
        float p   = o0a * o0b;
        float o1a = s_b1[k * 2 + 0] + s_w1[k * 26 + 12] * p;
        float o1b = s_b1[k * 2 + 1] + s_w1[k * 26 + 25] * p;
        #pragma unroll
        for (int j = 0; j < 12; ++j) {
            o1a += s_w1[k * 26 + j] * xv[j];
            o1b += s_w1[k * 26 + 13 + j] * xv[j];
        }
        #pragma unroll
        for (int d = 1; d <= 2; ++d) {
            int   j  = k * 6 + d;
            float dp = s_w0[k * 24 + j] * o0b + s_w0[k * 24 + 12 + j] * o0a;
            float d0 = s_w1[k * 26 + j] + s_w1[k * 26 + 12] * dp;
            float d1 = s_w1[k * 26 + 13 + j] + s_w1[k * 26 + 25] * dp;
            float dq = d0 * o1b + o1a * d1;
            float g  = s_wf[k * 14 + j] + s_wf[k * 14 + 12] * dp
                     + s_wf[k * 14 + 13] * dq;
            float alt = (d == 1) ? fe[k * 8 + 6] : fe[k * 8 + 7];
            xs[j] = (g > 0.0f) ? xv[j] : alt;
        }
    }

    // final forward polys on upwinded features (shared xs, per reference)
    float uadd[2];
    #pragma unroll
    for (int k = 0; k < 2; ++k) {
        float o0a = s_b0[k * 2 + 0], o0b = s_b0[k * 2 + 1];
        #pragma unroll
        for (int j = 0; j < 12; ++j) {
            o0a += s_w0[k * 24 + j] * xs[j];
            o0b += s_w0[k * 24 + 12 + j] * xs[j];
        }
        float p   = o0a * o0b;
        float o1a = s_b1[k * 2 + 0] + s_w1[k * 26 + 12] * p;
        float o1b = s_b1[k * 2 + 1] + s_w1[k * 26 + 25] * p;
        #pragma unroll
        for (int j = 0; j < 12; ++j) {
            o1a += s_w1[k * 26 + j] * xs[j];
            o1b += s_w1[k * 26 + 13 + j] * xs[j];
        }
        float q   = o1a * o1b;
        float out = s_bf[k] + s_wf[k * 14 + 12] * p + s_wf[k * 14 + 13] * q;
        #pragma unroll
        for (int j = 0; j < 12; ++j) out += s_wf[k * 14 + j] * xs[j];
        uadd[k] = out;
    }

    const int gy = ty0 + oy0 + hf;
    const int gx = tx0 + m;
    #pragma unroll
    for (int k = 0; k < 2; ++k) {
        size_t idx = (((size_t)b * 2 + k) * 192 + gy) * 192 + gx;
        dst[idx] = ubase[idx] + coef * uadd[k];
    }
}

extern "C" void kernel_launch(void* const* d_in, const int* in_sizes, int n_in,
                              void* d_out, int out_size, void* d_ws, size_t ws_size,
                              hipStream_t stream) {
    (void)in_sizes; (void)n_in; (void)out_size; (void)ws_size;
    const float* init = (const float*)d_in[0];
    const float* gk   = (const float*)d_in[1];
    const float* pw0  = (const float*)d_in[2];
    const float* pb0  = (const float*)d_in[3];
    const float* pw1  = (const float*)d_in[4];
    const float* pb1  = (const float*)d_in[5];
    const float* pwf  = (const float*)d_in[6];
    const float* pbf  = (const float*)d_in[7];
    // d_in[8] = T == 1 -> round(T/DT) = 5 RK2 steps (hardcoded).

    const size_t N = (size_t)16 * 2 * 192 * 192;
    float* u   = (float*)d_ws;
    float* uh  = u + N;
    float* out = (float*)d_out;

    dim3 grid(12, 12, 16), block(256);
    const float HC = PDT * 0.5f, FC = PDT;

    auto launch = [&](const float* s, const float* base, float* d, float c) {
        polypde_rhs_step<<<grid, block, 0, stream>>>(
            s, base, d, gk, pw0, pb0, pw1, pb1, pwf, pbf, c);
    };

    // step 1 (reads init, never mutates inputs)
    launch(init, init, uh, HC);
    launch(uh,   init, u,  FC);
    // steps 2..5 (pass B writes dst==base in-place: each thread only touches
    // its own pixel of base/dst, conv reads come from the other buffer)
    for (int st = 1; st < 5; ++st) {
        launch(u,  u, uh, HC);
        launch(uh, u, (st == 4) ? out : u, FC);
    }
}